// LinearAttentionWeights_31920196944161
// MI455X (gfx1250) — compile-verified
//
#include <hip/hip_runtime.h>
#include <hip/hip_bf16.h>

typedef __attribute__((ext_vector_type(2))) float v2f;
typedef __attribute__((ext_vector_type(4))) float v4f;
typedef __attribute__((ext_vector_type(8))) float v8f;

#define LA_B 2
#define LA_H 16
#define LA_L 2048
#define LA_D 64
#define LA_NT (LA_L / 16)   // 128 column/row tiles

// ---------------------------------------------------------------------------
// Kernel 1: per-head linear feature maps via fp32 WMMA (16x16x4).
//   q_f[b,h,l,e] = sum_d Q[b,h,l,d] * w_fm_k[h,d,e]   (blockIdx.y == 0)
//   k_f[b,h,l,e] = sum_d K[b,h,l,d] * w_fm_q[h,d,e]   (blockIdx.y == 1)
// 4 waves per block; wave w owns output column tile e0 = 16*w (D=64 -> 4 tiles).
// ---------------------------------------------------------------------------
__global__ __launch_bounds__(128) void cdna5_feature_map(
    const float* __restrict__ q, const float* __restrict__ k,
    const float* __restrict__ w_fm_q, const float* __restrict__ w_fm_k,
    float* __restrict__ q_f, float* __restrict__ k_f)
{
    const int bh = blockIdx.x / LA_NT;
    const int it = blockIdx.x % LA_NT;
    const int h  = bh % LA_H;

    const float* __restrict__ src = (blockIdx.y == 0) ? q : k;
    // NOTE: reference applies feature_map_k to queries and feature_map_q to keys.
    const float* __restrict__ W =
        ((blockIdx.y == 0) ? w_fm_k : w_fm_q) + (size_t)h * LA_D * LA_D;
    float* __restrict__ dst = (blockIdx.y == 0) ? q_f : k_f;

    const int wave = threadIdx.x >> 5;
    const int lane = threadIdx.x & 31;
    const int hi = lane >> 4;      // lane half selects K pair / row half
    const int lo = lane & 15;      // M index for A, N index for B/C
    const int r0 = it * 16;
    const int e0 = wave * 16;

    const float* __restrict__ srcBH = src + (size_t)bh * LA_L * LA_D;

    v8f c = {};
#pragma unroll
    for (int k0 = 0; k0 < LA_D; k0 += 4) {
        // A[lo][k0 + 2*hi + j] : contiguous pair in row-major src
        v2f a = *(const v2f*)(srcBH + (size_t)(r0 + lo) * LA_D + k0 + 2 * hi);
        // B[k][lo] = W[k][e0+lo], k = k0 + 2*hi + j : stride-D apart -> 2 loads
        v2f b;
        b.x = W[(size_t)(k0 + 2 * hi + 0) * LA_D + e0 + lo];
        b.y = W[(size_t)(k0 + 2 * hi + 1) * LA_D + e0 + lo];
        c = __builtin_amdgcn_wmma_f32_16x16x4_f32(
                /*neg_a=*/false, a, /*neg_b=*/false, b,
                /*c_mod=*/(short)0, c, /*reuse_a=*/false, /*reuse_b=*/false);
    }

    float* __restrict__ dstBH = dst + (size_t)bh * LA_L * LA_D;
#pragma unroll
    for (int v = 0; v < 8; ++v) {
        const int row = r0 + v + 8 * hi;        // C layout: VGPR v, lane half hi
        dstBH[(size_t)row * LA_D + e0 + lo] = c[v];
    }
}

// ---------------------------------------------------------------------------
// Kernel 2a: inclusive prefix sum of k_f along L, shifted for the k=1 causal
// diagonal:  S[i] = sum_{j=0}^{min(i+1, L-1)} k_f[j]   (per (b,h), per d).
// One 64-thread block per (b,h); thread d walks L (loads are coalesced per i).
// ---------------------------------------------------------------------------
__global__ __launch_bounds__(64) void cdna5_cumsum_kf(
    const float* __restrict__ k_f, float* __restrict__ S)
{
    const int bh = blockIdx.x;
    const int d  = threadIdx.x;
    const float* __restrict__ kf = k_f + (size_t)bh * LA_L * LA_D + d;
    float* __restrict__ so       = S   + (size_t)bh * LA_L * LA_D + d;

    float s = 0.0f;
#pragma unroll 4
    for (int i = 0; i < LA_L; ++i) {
        s += kf[(size_t)i * LA_D];
        if (i >= 1) so[(size_t)(i - 1) * LA_D] = s;   // S[i-1] = cumsum through i
    }
    so[(size_t)(LA_L - 1) * LA_D] = s;                // S[L-1] = cumsum through L-1
}

// ---------------------------------------------------------------------------
// Kernel 2b: rowsum[i] = q_f[i] . S[i];  store reciprocal for fused normalize.
// ---------------------------------------------------------------------------
__global__ __launch_bounds__(256) void cdna5_rowsum_inv(
    const float* __restrict__ q_f, const float* __restrict__ S,
    float* __restrict__ inv)
{
    const int r = blockIdx.x * 256 + threadIdx.x;     // 0 .. B*H*L-1
    const v4f* __restrict__ qp = (const v4f*)(q_f + (size_t)r * LA_D);
    const v4f* __restrict__ sp = (const v4f*)(S   + (size_t)r * LA_D);
    float acc = 0.0f;
#pragma unroll
    for (int d4 = 0; d4 < LA_D / 4; ++d4) {
        const v4f a = qp[d4];
        const v4f b = sp[d4];
        acc += a.x * b.x + a.y * b.y + a.z * b.z + a.w * b.w;
    }
    inv[r] = 1.0f / acc;
}

// ---------------------------------------------------------------------------
// Kernel 3: attn tile strip.  One block = one (b,h, 16-row strip).
// Wave w computes column tiles jt in {w, w+4, w+8, ...} up to the causal
// limit (jt <= it+1), TWO tiles per loop iteration with independent WMMA
// accumulation chains (shared A fragments) for ILP.  The second tile's
// column base is clamped instead of branched so EXEC stays all-ones around
// every WMMA; only its store is guarded.  The col <= row+1 mask and the
// 1/rowsum scale are fused into the store.  All output stores are
// NON-TEMPORAL: the 512 MB output is write-once, and RT stores would evict
// the hot q_f/k_f working set from the 192 MB L2.
// ---------------------------------------------------------------------------
__global__ __launch_bounds__(128) void cdna5_attn(
    const float* __restrict__ q_f, const float* __restrict__ k_f,
    const float* __restrict__ inv, float* __restrict__ out)
{
    const int bh = blockIdx.x / LA_NT;
    const int it = blockIdx.x % LA_NT;
    const int wave = threadIdx.x >> 5;
    const int lane = threadIdx.x & 31;
    const int hi = lane >> 4;
    const int lo = lane & 15;
    const int r0 = it * 16;

    const float* __restrict__ qBH = q_f + (size_t)bh * LA_L * LA_D;
    const float* __restrict__ kBH = k_f + (size_t)bh * LA_L * LA_D;
    float* __restrict__ oBH       = out + (size_t)bh * LA_L * LA_L;

    // Preload all 16 A fragments for this row strip (reused for every col tile).
    v2f afrag[16];
#pragma unroll
    for (int s = 0; s < 16; ++s)
        afrag[s] = *(const v2f*)(qBH + (size_t)(r0 + lo) * LA_D + 4 * s + 2 * hi);

    // Preload reciprocal row sums for this lane's 8 rows.
    float invv[8];
#pragma unroll
    for (int v = 0; v < 8; ++v)
        invv[v] = inv[(size_t)bh * LA_L + r0 + v + 8 * hi];

    const int jmax = min(it + 1, LA_NT - 1);   // last tile touching the band
    for (int jt = wave; jt <= jmax; jt += 8) { // wave-uniform: EXEC stays full
        const int jt2  = min(jt + 4, jmax);    // clamped (may duplicate jt's work;
                                               // avoids branching around WMMA)
        const int c0a = jt * 16;
        const int c0b = jt2 * 16;
        v8f ca = {};
        v8f cb = {};
#pragma unroll
        for (int s = 0; s < 16; ++s) {
            // B[k][lo] = k_f[c0+lo][4s + 2*hi + j] : contiguous pair (free transpose)
            v2f b0 = *(const v2f*)(kBH + (size_t)(c0a + lo) * LA_D + 4 * s + 2 * hi);
            v2f b1 = *(const v2f*)(kBH + (size_t)(c0b + lo) * LA_D + 4 * s + 2 * hi);
            ca = __builtin_amdgcn_wmma_f32_16x16x4_f32(
                     false, afrag[s], false, b0, (short)0, ca, false, false);
            cb = __builtin_amdgcn_wmma_f32_16x16x4_f32(
                     false, afrag[s], false, b1, (short)0, cb, false, false);
        }
#pragma unroll
        for (int v = 0; v < 8; ++v) {
            const int row = r0 + v + 8 * hi;
            const int col = c0a + lo;
            const float val = (col <= row + 1) ? ca[v] * invv[v] : 0.0f;
            __builtin_nontemporal_store(val, oBH + (size_t)row * LA_L + col);
        }
        if (jt + 4 <= jmax) {                  // store guard only (no WMMA inside)
#pragma unroll
            for (int v = 0; v < 8; ++v) {
                const int row = r0 + v + 8 * hi;
                const int col = c0b + lo;
                const float val = (col <= row + 1) ? cb[v] * invv[v] : 0.0f;
                __builtin_nontemporal_store(val, oBH + (size_t)row * LA_L + col);
            }
        }
    }

    // Zero-fill the strictly-causal-masked region (tiles jt >= it+2).
    const int zstart = (it + 2) * 16;
    if (zstart < LA_L) {
        const int nz4 = (LA_L - zstart) >> 2;          // v4f's per row
        const v4f z4 = {0.0f, 0.0f, 0.0f, 0.0f};
        for (int idx = threadIdx.x; idx < 16 * nz4; idx += 128) {
            const int row = r0 + idx / nz4;
            const int c4  = idx % nz4;
            __builtin_nontemporal_store(
                z4, (v4f*)(oBH + (size_t)row * LA_L + zstart + 4 * c4));
        }
    }
}

// ---------------------------------------------------------------------------
// Host launcher.  Workspace layout (floats):
//   q_f [B*H*L*D] | k_f [B*H*L*D] | S [B*H*L*D] | inv [B*H*L]   (~50.6 MB)
// ---------------------------------------------------------------------------
extern "C" void kernel_launch(void* const* d_in, const int* in_sizes, int n_in,
                              void* d_out, int out_size, void* d_ws, size_t ws_size,
                              hipStream_t stream) {
    const float* q     = (const float*)d_in[0];   // query_states [B,H,L,D]
    const float* k     = (const float*)d_in[1];   // key_states   [B,H,L,D]
    const float* w_fmq = (const float*)d_in[2];   // w_fm_q [H,D,D]
    const float* w_fmk = (const float*)d_in[3];   // w_fm_k [H,D,D]
    float* out = (float*)d_out;

    const size_t nQF = (size_t)LA_B * LA_H * LA_L * LA_D;
    float* ws  = (float*)d_ws;
    float* q_f = ws;
    float* k_f = q_f + nQF;
    float* S   = k_f + nQF;
    float* inv = S + nQF;

    dim3 g1(LA_B * LA_H * LA_NT, 2);
    cdna5_feature_map<<<g1, 128, 0, stream>>>(q, k, w_fmq, w_fmk, q_f, k_f);

    cdna5_cumsum_kf<<<LA_B * LA_H, 64, 0, stream>>>(k_f, S);

    cdna5_rowsum_inv<<<(LA_B * LA_H * LA_L) / 256, 256, 0, stream>>>(q_f, S, inv);

    cdna5_attn<<<LA_B * LA_H * LA_NT, 128, 0, stream>>>(q_f, k_f, inv, out);
}